// CRNNLayer_9981503996033
// MI455X (gfx1250) — compile-verified
//
#include <hip/hip_runtime.h>
#include <hip/hip_bf16.h>
#include <math.h>

// CRNN: im2col + 3-layer stacked LSTM scan.
// B=32, T=2048, C=128, K=8, S=4 -> P=511 ; KC=1024 ; H=256 ; 4H=1024.

typedef __attribute__((ext_vector_type(2))) float v2f;
typedef __attribute__((ext_vector_type(8))) float v8f;

static __device__ __forceinline__ v8f wmma4(v2f a, v2f b, v8f c) {
    // V_WMMA_F32_16X16X4_F32 : D = A(16x4) * B(4x16) + C(16x16), fp32 throughout
    return __builtin_amdgcn_wmma_f32_16x16x4_f32(false, a, false, b, (short)0, c,
                                                 false, false);
}

static __device__ __forceinline__ float sigmoidf_(float x) {
    return 1.0f / (1.0f + __expf(-x));
}

// ---------------------------------------------------------------------------
// Repack a [rows][1024] matrix into WMMA-B pair layout [rows/2][1024][2]:
//   out[(k>>1)*2048 + 2n + (k&1)] = in[k*1024 + n]
// so each lane's B operand {B[k][n], B[k+1][n]} is one contiguous b64 load.
// ---------------------------------------------------------------------------
__global__ __launch_bounds__(256) void pack_pairs(
    const float* __restrict__ in, float* __restrict__ out, int total /* =(rows/2)*1024 */)
{
    const int idx = blockIdx.x * 256 + threadIdx.x;
    if (idx >= total) return;
    const int pair = idx >> 10;
    const int n    = idx & 1023;
    out[2 * idx]     = in[(2 * pair) * 1024 + n];
    out[2 * idx + 1] = in[(2 * pair + 1) * 1024 + n];
}

// ---------------------------------------------------------------------------
// Kernel A: Z0[p][b][n] = im2col(x)[p,b,:] @ W0[:,n] + b0[n]
//   rows m = p*32 + b (M = 16352 = 1022 tiles), K = 1024, N = 1024 (64 tiles).
//   The K*C patch window is CONTIGUOUS in x (row stride = C), so the A operand
//   is just xbase + k : pure pointer induction, no per-step address math.
//   Each wave computes 4 N-tiles sharing one A load: 4 wmma per 5 loads.
// ---------------------------------------------------------------------------
__global__ __launch_bounds__(256) void crnn_im2col_gemm(
    const float* __restrict__ x,      // [32][2048][128]
    const float* __restrict__ W0pk,   // [512][1024][2]
    const float* __restrict__ b0,     // [1024]
    float* __restrict__ Z0)           // [511][32][1024]
{
    const int lane   = threadIdx.x & 31;
    const int wave   = threadIdx.x >> 5;
    const int laneLo = lane & 15;
    const int laneHi = lane >> 4;              // 0: K0/K1 half, 1: K2/K3 half

    const int mtile = blockIdx.x;              // 0..1021
    const int nb    = (blockIdx.y * 8 + wave) * 4;  // first of 4 ntiles (0..60)

    const int m = mtile * 16 + laneLo;         // patch row (same for both halves)
    const int p = m >> 5;                      // patch index
    const int b = m & 31;                      // batch index

    // patch(m, k) == x[(b*2048 + p*4)*128 + k]  (contiguous window)
    const float* xa = x + (size_t)(b * 2048 + p * 4) * 128 + laneHi * 2;
    const float* wb = W0pk + (size_t)laneHi * 2048 + 2 * (nb * 16 + laneLo);

    v8f accs[4] = {v8f{}, v8f{}, v8f{}, v8f{}};
#pragma unroll 4
    for (int k0 = 0; k0 < 1024; k0 += 4) {
        v2f a = *(const v2f*)(xa + k0);                    // one b64 A load
        const float* bb = wb + (size_t)(k0 >> 1) * 2048;   // 4 b64 B loads, imm offs
        accs[0] = wmma4(a, *(const v2f*)(bb),      accs[0]);
        accs[1] = wmma4(a, *(const v2f*)(bb + 32), accs[1]);
        accs[2] = wmma4(a, *(const v2f*)(bb + 64), accs[2]);
        accs[3] = wmma4(a, *(const v2f*)(bb + 96), accs[3]);
    }

#pragma unroll
    for (int g = 0; g < 4; ++g) {
        const int n = (nb + g) * 16 + laneLo;
        const float bias = b0[n];
#pragma unroll
        for (int r = 0; r < 8; ++r) {
            const int rowD = mtile * 16 + laneHi * 8 + r;  // D layout: r / r+8
            Z0[rowD * 1024 + n] = accs[g][r] + bias;
        }
    }
}

// ---------------------------------------------------------------------------
// One [32x256] @ [256x1024] accumulation into the 4 gate tiles.
//   hPb : LDS packed h base (already offset by laneHi*64 + mA*2)
//   Bb  : packed weights base (already offset by laneHi*2048 + 2*j);
//         gates at +512/+1024/+1536 floats (2/4/6 KB immediate offsets)
// ---------------------------------------------------------------------------
static __device__ __forceinline__ void mm_accum(
    const float* __restrict__ hPb, const float* __restrict__ Bb,
    v8f& ai, v8f& af, v8f& ag, v8f& ao)
{
#pragma unroll 4
    for (int k0 = 0; k0 < 256; k0 += 4) {
        v2f a = *(const v2f*)(hPb + (k0 >> 1) * 64);            // ds_load_b64
        const float* base = Bb + (size_t)(k0 >> 1) * 2048;      // global b64 x4
        ai = wmma4(a, *(const v2f*)(base),        ai);
        af = wmma4(a, *(const v2f*)(base + 512),  af);
        ag = wmma4(a, *(const v2f*)(base + 1024), ag);
        ao = wmma4(a, *(const v2f*)(base + 1536), ao);
    }
}

// ---------------------------------------------------------------------------
// Kernel B: persistent single-workgroup LSTM scan (1024 threads = 32 waves).
//   LDS: hP[3][128(k/2)][32(b)][2] pair-packed (96 KB). c-state in registers.
//   Wave w = (slice s = w>>1, mtile = w&1) computes gate tiles i/f/g/o for
//   hidden units j in [16s, 16s+16) and batch rows [16*mtile, 16*mtile+16).
//   Each lane ends up holding i,f,g,o of the same (b,j) -> fused gate math.
// ---------------------------------------------------------------------------
__global__ __launch_bounds__(1024) void crnn_lstm_scan(
    const float* __restrict__ Z0,    // [511][32][1024]  (x@W0 + b0)
    const float* __restrict__ U0pk,
    const float* __restrict__ W1pk, const float* __restrict__ U1pk,
    const float* __restrict__ b1,
    const float* __restrict__ W2pk, const float* __restrict__ U2pk,
    const float* __restrict__ b2,
    float* __restrict__ out)         // [32][511][256]
{
    extern __shared__ float sh[];    // hP: 3 * 8192 floats = 96 KB

    const int tid    = threadIdx.x;
    const int lane   = tid & 31;
    const int wave   = tid >> 5;
    const int laneLo = lane & 15;
    const int laneHi = lane >> 4;

    const int s     = wave >> 1;            // hidden-unit slice 0..15
    const int mtile = wave & 1;             // batch-row tile 0..1
    const int mA    = mtile * 16 + laneLo;  // A-operand batch row
    const int j     = s * 16 + laneLo;      // hidden unit owned by lane
    const int bBase = mtile * 16 + laneHi * 8;  // D rows bBase..bBase+7
    const int jPair = (j >> 1) * 64 + (j & 1);  // packed h write base for unit j
    const int aOff  = laneHi * 64 + mA * 2;     // per-lane A-operand LDS offset
    const int bOff  = laneHi * 2048 + 2 * j;    // per-lane B-operand base offset

    // zero initial hidden state
    for (int i = tid; i < 3 * 8192; i += 1024) sh[i] = 0.0f;
    float cst[3][8];
#pragma unroll
    for (int l = 0; l < 3; ++l)
#pragma unroll
        for (int r = 0; r < 8; ++r) cst[l][r] = 0.0f;
    __syncthreads();

    const float* Wl[3] = {nullptr, W1pk, W2pk};
    const float* Ul[3] = {U0pk, U1pk, U2pk};
    const float* bl[3] = {nullptr, b1, b2};

    for (int t = 0; t < 511; ++t) {
#pragma unroll
        for (int l = 0; l < 3; ++l) {
            v8f ai, af, ag, ao;
            if (l == 0) {
                // z init = precomputed x@W0 + b0
#pragma unroll
                for (int r = 0; r < 8; ++r) {
                    const float* zr = Z0 + (size_t)(t * 32 + bBase + r) * 1024;
                    ai[r] = zr[j];
                    af[r] = zr[256 + j];
                    ag[r] = zr[512 + j];
                    ao[r] = zr[768 + j];
                }
            } else {
                const float bi = bl[l][j],       bf = bl[l][256 + j];
                const float bg = bl[l][512 + j], bo = bl[l][768 + j];
#pragma unroll
                for (int r = 0; r < 8; ++r) { ai[r] = bi; af[r] = bf; ag[r] = bg; ao[r] = bo; }
            }

            // layers 1,2: + h_{l-1}(t) @ W_l   (A from LDS, current step)
            if (l > 0)
                mm_accum(sh + (l - 1) * 8192 + aOff, Wl[l] + bOff, ai, af, ag, ao);

            // + h_l(t-1) @ U_l   (A from LDS, previous step)
            mm_accum(sh + l * 8192 + aOff, Ul[l] + bOff, ai, af, ag, ao);

            // fused gates: each lane holds i,f,g,o for its own (b, j) cells
            float hval[8];
#pragma unroll
            for (int r = 0; r < 8; ++r) {
                const float iv = sigmoidf_(ai[r]);
                const float fv = sigmoidf_(af[r]);
                const float gv = tanhf(ag[r]);
                const float ov = sigmoidf_(ao[r]);
                const float cn = fv * cst[l][r] + iv * gv;
                cst[l][r] = cn;
                hval[r] = ov * tanhf(cn);
            }

            __syncthreads();   // all reads of hP[l] / hP[l-1] done
            {
                float* hme = sh + l * 8192;
#pragma unroll
                for (int r = 0; r < 8; ++r)
                    hme[jPair + (bBase + r) * 2] = hval[r];
            }
            __syncthreads();   // h_l(t) visible for next layer / step

            if (l == 2) {
#pragma unroll
                for (int r = 0; r < 8; ++r)
                    out[((size_t)(bBase + r) * 511 + t) * 256 + j] = hval[r];
            }
        }
    }
}

// ---------------------------------------------------------------------------
extern "C" void kernel_launch(void* const* d_in, const int* in_sizes, int n_in,
                              void* d_out, int out_size, void* d_ws, size_t ws_size,
                              hipStream_t stream) {
    const float* x  = (const float*)d_in[0];
    const float* W0 = (const float*)d_in[1];
    const float* U0 = (const float*)d_in[2];
    const float* b0 = (const float*)d_in[3];
    const float* W1 = (const float*)d_in[4];
    const float* U1 = (const float*)d_in[5];
    const float* b1 = (const float*)d_in[6];
    const float* W2 = (const float*)d_in[7];
    const float* U2 = (const float*)d_in[8];
    const float* b2 = (const float*)d_in[9];
    float* out = (float*)d_out;

    // Workspace layout (floats):
    //   Z0   : 511*32*1024           = 16,744,448
    //   W0pk : 1024*1024             =  1,048,576
    //   U0pk,W1pk,U1pk,W2pk,U2pk     =    262,144 each
    float* Z0   = (float*)d_ws;
    float* W0pk = Z0 + (size_t)511 * 32 * 1024;
    float* U0pk = W0pk + 1024 * 1024;
    float* W1pk = U0pk + 256 * 1024;
    float* U1pk = W1pk + 256 * 1024;
    float* W2pk = U1pk + 256 * 1024;
    float* U2pk = W2pk + 256 * 1024;

    // Phase 0: repack all weight matrices into WMMA-B pair layout
    pack_pairs<<<2048, 256, 0, stream>>>(W0, W0pk, 512 * 1024);
    pack_pairs<<<512, 256, 0, stream>>>(U0, U0pk, 128 * 1024);
    pack_pairs<<<512, 256, 0, stream>>>(W1, W1pk, 128 * 1024);
    pack_pairs<<<512, 256, 0, stream>>>(U1, U1pk, 128 * 1024);
    pack_pairs<<<512, 256, 0, stream>>>(W2, W2pk, 128 * 1024);
    pack_pairs<<<512, 256, 0, stream>>>(U2, U2pk, 128 * 1024);

    // Phase 1: whole-chip im2col GEMM (recurrence-independent part)
    crnn_im2col_gemm<<<dim3(1022, 2), 256, 0, stream>>>(x, W0pk, b0, Z0);

    // Phase 2: persistent single-WGP LSTM scan, 96 KB dynamic LDS for h-state
    crnn_lstm_scan<<<1, 1024, 3 * 8192 * sizeof(float), stream>>>(
        Z0, U0pk, W1pk, U1pk, b1, W2pk, U2pk, b2, out);
}